// BandFunctionalPrior_89395449299813
// MI455X (gfx1250) — compile-verified
//
#include <hip/hip_runtime.h>
#include <math.h>

typedef __attribute__((ext_vector_type(2))) float v2f;
typedef __attribute__((ext_vector_type(8))) float v8f;

#define BATCH 32
#define NBANDS 5
#define NC 64
#define NT 128
#define DEMB 64
#define DS 16
#define PFUNC_ELEMS (BATCH * NT * NC * NC)   // 16,777,216

// ---------------------------------------------------------------------------
// Prep kernel: per-band projected vectors u,v,p,r (16 floats each) + adj bitmask
//   u = Wq * W_e[n], v = Wk * W_e[n], p = Wq * b_e[n], r = Wk * b_e[n]
// uvpr layout: [n][typ][s], typ: 0=u,1=v,2=p,3=r  -> 5*4*16 = 320 floats
// amask: 64 rows * 2 words (uint32 bitmask of adj>0)
// ---------------------------------------------------------------------------
__global__ void bfp_prep_kernel(const float* __restrict__ adj,
                                const float* __restrict__ W_e,
                                const float* __restrict__ b_e,
                                const float* __restrict__ Wq,
                                const float* __restrict__ Wk,
                                float* __restrict__ uvpr,
                                unsigned* __restrict__ amask) {
  int tid = threadIdx.x;
  for (int idx = tid; idx < NBANDS * 64; idx += blockDim.x) {
    int n = idx >> 6;
    int rem = idx & 63;
    int typ = rem >> 4;
    int s = rem & 15;
    const float* M = (typ == 0 || typ == 2) ? Wq : Wk;   // u,p use Wq; v,r use Wk
    const float* V = (typ < 2) ? W_e : b_e;              // u,v use W_e; p,r use b_e
    float acc = 0.0f;
#pragma unroll 8
    for (int d = 0; d < DEMB; ++d) acc += M[s * DEMB + d] * V[n * DEMB + d];
    uvpr[idx] = acc;
  }
  for (int idx = tid; idx < NC * 2; idx += blockDim.x) {
    int row = idx >> 1, half = idx & 1;
    unsigned m = 0;
    for (int bit = 0; bit < 32; ++bit)
      if (adj[row * NC + half * 32 + bit] > 0.0f) m |= (1u << bit);
    amask[idx] = m;
  }
}

// ---------------------------------------------------------------------------
// Main kernel: one block = one (b, t). 256 threads = 8 wave32.
// Per band: s = q k^T via v_wmma_f32_16x16x4_f32 (each wave: 2 tiles x 4 K-steps),
// masked row softmax, weighted accumulation over bands in registers.
// ---------------------------------------------------------------------------
__global__ void __launch_bounds__(256)
bfp_main_kernel(const float* __restrict__ x,
                const float* __restrict__ amask_in,
                const float* __restrict__ aalpha,
                const float* __restrict__ W_e,
                const float* __restrict__ b_e,
                const float* __restrict__ uvpr,
                const unsigned* __restrict__ adjmask,
                float* __restrict__ out) {
  __shared__ float sx[NBANDS][NC];     // x[b, n, :, t]
  __shared__ float suv[NBANDS * 64];   // u,v,p,r per band
  __shared__ unsigned smask[NC * 2];   // adj row bitmasks
  __shared__ float sw[NBANDS];         // alpha * mask
  __shared__ float sm5[NBANDS];        // mask
  __shared__ float sxb[NBANDS];        // mean_c x
  __shared__ float sbuf[NC][NC + 1];   // one band's score matrix (scaled)

  const int bid = blockIdx.x;          // 0..4095
  const int b = bid >> 7;              // / NT
  const int t = bid & (NT - 1);
  const int tid = threadIdx.x;
  const int wave = tid >> 5;
  const int lane = tid & 31;

  // ---- stage 0: cooperative loads ----
  for (int idx = tid; idx < NBANDS * 64; idx += 256) {
    int n = idx >> 6, c = idx & 63;
    sx[n][c] = x[(((b * NBANDS + n) << 6) + c) * NT + t];
    suv[idx] = uvpr[idx];
  }
  for (int idx = tid; idx < NC * 2; idx += 256) smask[idx] = adjmask[idx];
  if (tid < NBANDS) {
    float mm = amask_in[(b * NT + t) * NBANDS + tid];
    float aa = aalpha[(b * NT + t) * NBANDS + tid];
    sm5[tid] = mm;
    sw[tid] = mm * aa;
  }
  __syncthreads();

  // softmax-stage thread mapping: 4 threads per row, 16 cols each
  const int row_i = tid >> 2;
  const int jc = (tid & 3) << 4;
  const unsigned mword = smask[row_i * 2 + (jc >> 5)] >> (jc & 31);

  // WMMA-stage tile mapping: wave -> row-tile I, col-tiles Jbase, Jbase+1
  const int I = wave >> 1;
  const int Jbase = (wave & 1) * 2;
  const int half = lane >> 4;   // lane half selects K pair {0,1} vs {2,3}
  const int lm = lane & 15;

  float accv[16];
#pragma unroll
  for (int k = 0; k < 16; ++k) accv[k] = 0.0f;

  for (int n = 0; n < NBANDS; ++n) {
    const float* u = &suv[n * 64 + 0];
    const float* v = &suv[n * 64 + 16];
    const float* p = &suv[n * 64 + 32];
    const float* r = &suv[n * 64 + 48];
    const float xa = sx[n][16 * I + lm];   // A-row electrode value

    // ---- s = q k^T via WMMA (f32 16x16x4), 2 tiles per wave ----
#pragma unroll
    for (int Jo = 0; Jo < 2; ++Jo) {
      const int J = Jbase + Jo;
      const float xj = sx[n][16 * J + lm]; // B-col electrode value
      v8f cacc = {};
#pragma unroll
      for (int k4 = 0; k4 < DS; k4 += 4) {
        const int s0 = k4 + (half << 1);
        v2f a, bb;
        a.x = xa * u[s0] + p[s0];          // q[16I+lm][s0]
        a.y = xa * u[s0 + 1] + p[s0 + 1];  // q[16I+lm][s0+1]
        bb.x = xj * v[s0] + r[s0];         // k[16J+lm][s0]
        bb.y = xj * v[s0 + 1] + r[s0 + 1]; // k[16J+lm][s0+1]
        cacc = __builtin_amdgcn_wmma_f32_16x16x4_f32(
            false, a, false, bb, (short)0, cacc, false, false);
      }
      // D layout: VGPR m, lanes0-15 -> M=m, lanes16-31 -> M=m+8; N = lane%16
#pragma unroll
      for (int m = 0; m < 8; ++m)
        sbuf[16 * I + half * 8 + m][16 * J + lm] = cacc[m] * 0.25f; // /sqrt(d_s)
    }
    __syncthreads();

    // ---- masked row softmax + band-weighted accumulation ----
    float vals[16];
    const float* srow = &sbuf[row_i][jc];
#pragma unroll
    for (int k = 0; k < 16; ++k) vals[k] = srow[k];

    float mx = -INFINITY;
#pragma unroll
    for (int k = 0; k < 16; ++k)
      if ((mword >> k) & 1u) mx = fmaxf(mx, vals[k]);
    // row group = 4 consecutive lanes of the same wave
    mx = fmaxf(mx, __shfl_xor(mx, 1, 32));
    mx = fmaxf(mx, __shfl_xor(mx, 2, 32));

    float ssum = 0.0f;
#pragma unroll
    for (int k = 0; k < 16; ++k) {
      float e = ((mword >> k) & 1u) ? __expf(vals[k] - mx) : 0.0f;
      vals[k] = e;
      ssum += e;
    }
    ssum += __shfl_xor(ssum, 1, 32);
    ssum += __shfl_xor(ssum, 2, 32);

    const float scale = sw[n] / ssum;
#pragma unroll
    for (int k = 0; k < 16; ++k) accv[k] += vals[k] * scale;
    __syncthreads();  // sbuf reused by next band
  }

  // ---- write P_func (coalesced float4) ----
  float* outP = out + (size_t)bid * (NC * NC) + row_i * NC + jc;
#pragma unroll
  for (int k = 0; k < 16; k += 4) {
    float4 v4 = {accv[k], accv[k + 1], accv[k + 2], accv[k + 3]};
    *(float4*)(outP + k) = v4;
  }

  // ---- embeddings_mean ----
  if (tid < NBANDS) {
    float s = 0.0f;
#pragma unroll 8
    for (int c = 0; c < NC; ++c) s += sx[tid][c];
    sxb[tid] = s * (1.0f / NC);
  }
  __syncthreads();
  if (tid < DEMB) {
    float acc = 0.0f, cnt = 0.0f;
#pragma unroll
    for (int n = 0; n < NBANDS; ++n) {
      float m = sm5[n];
      cnt += m;
      acc += m * (sxb[n] * W_e[n * DEMB + tid] + b_e[n * DEMB + tid]);
    }
    cnt = fmaxf(cnt, 1.0f);
    out[PFUNC_ELEMS + (size_t)bid * DEMB + tid] = acc / cnt;
  }
}

// ---------------------------------------------------------------------------
extern "C" void kernel_launch(void* const* d_in, const int* in_sizes, int n_in,
                              void* d_out, int out_size, void* d_ws, size_t ws_size,
                              hipStream_t stream) {
  const float* x = (const float*)d_in[0];       // (32,5,64,128)
  const float* adj = (const float*)d_in[1];     // (64,64)
  const float* amsk = (const float*)d_in[2];    // (32,128,5)
  const float* aalp = (const float*)d_in[3];    // (32,128,5)
  const float* W_e = (const float*)d_in[4];     // (5,64)
  const float* b_e = (const float*)d_in[5];     // (5,64)
  const float* Wq = (const float*)d_in[6];      // (16,64)
  const float* Wk = (const float*)d_in[7];      // (16,64)
  float* out = (float*)d_out;

  float* uvpr = (float*)d_ws;                       // 320 floats
  unsigned* amaskws = (unsigned*)(uvpr + 320);      // 128 words

  hipLaunchKernelGGL(bfp_prep_kernel, dim3(1), dim3(256), 0, stream,
                     adj, W_e, b_e, Wq, Wk, uvpr, amaskws);
  hipLaunchKernelGGL(bfp_main_kernel, dim3(BATCH * NT), dim3(256), 0, stream,
                     x, amsk, aalp, W_e, b_e, uvpr, amaskws, out);
}